// APRConvNet_61443802136698
// MI455X (gfx1250) — compile-verified
//
#include <hip/hip_runtime.h>

// ---------------------------------------------------------------------------
// APRConvNet for MI455X (gfx1250, wave32).
// Layer2 (dominant FLOPs) runs on V_WMMA_F32_16X16X4_F32 (fp32-exact).
// B operand (W2) held in LDS interleaved by K-row pairs so each WMMA B
// fragment is a single ds_load_b64 (no register shuffling).
// ---------------------------------------------------------------------------

#define N1C 1000000
#define N2C 125000
#define N3C 15625
#define KN  27

typedef float v2f __attribute__((ext_vector_type(2)));
typedef float v8f __attribute__((ext_vector_type(8)));

// float atomic max via int/uint atomics; valid with -inf initialization.
__device__ __forceinline__ void atomicMaxF(float* addr, float v) {
  if (v >= 0.0f)
    atomicMax(reinterpret_cast<int*>(addr), __float_as_int(v));
  else
    atomicMin(reinterpret_cast<unsigned int*>(addr), __float_as_uint(v));
}

// ---------------------------------------------------------------------------
// Workspace layout (floats):
//   [0, 2,000,000)            h1pool  [N2,16]   (segment-max accum, init -inf)
//   [2,000,000, 2,500,000)    h2pool  [N3,32]   (segment-max accum, init -inf)
//   [2,500,000, 2,500,256)    stats: sums1[16] sqs1[16] scale1[16] shift1[16]
//                                    sums2[32] sqs2[32] scale2[32] shift2[32]
//                                    logit_acc[10] (zero-init)
//   [2,500,256, 4,500,256)    h1n     [N2,16]   (post-BN activations)
// Total ~17.2 MB.
// ---------------------------------------------------------------------------

__global__ void init_ws_kernel(float* __restrict__ ws) {
  size_t i = (size_t)blockIdx.x * blockDim.x + threadIdx.x;
  const size_t POOLS = 2500000;            // h1pool + h2pool
  const size_t STATS = 256;
  if (i < POOLS)              ws[i] = -__builtin_inff();
  else if (i < POOLS + STATS) ws[i] = 0.0f;
}

// ---- Layer 1: multi-stencil conv [N1,1] -> [N1,16] fused with segment-max --
__global__ void __launch_bounds__(256)
layer1_kernel(const float* __restrict__ x, const float* __restrict__ W1,
              const float* __restrict__ b1, const int* __restrict__ nbr1,
              const int* __restrict__ stencil1, const int* __restrict__ pool1,
              float* __restrict__ h1pool, int n1) {
  __shared__ float w[2 * KN * 16];
  __shared__ float bb[16];
  for (int i = threadIdx.x; i < 2 * KN * 16; i += blockDim.x) w[i] = W1[i];
  for (int i = threadIdx.x; i < 16; i += blockDim.x) bb[i] = b1[i];
  __syncthreads();

  int nid = blockIdx.x * blockDim.x + threadIdx.x;
  if (nid >= n1) return;

  const float* Ws = &w[stencil1[nid] * (KN * 16)];
  float acc[16];
  #pragma unroll
  for (int d = 0; d < 16; ++d) acc[d] = bb[d];

  const int* nb = nbr1 + (size_t)nid * KN;
  for (int k = 0; k < KN; ++k) {
    float xv = x[nb[k]];
    const float* wk = Ws + k * 16;
    #pragma unroll
    for (int d = 0; d < 16; ++d) acc[d] = fmaf(xv, wk[d], acc[d]);
  }

  float* dst = h1pool + (size_t)pool1[nid] * 16;
  #pragma unroll
  for (int d = 0; d < 16; ++d) atomicMaxF(dst + d, acc[d]);
}

// ---- BatchNorm stats over relu(pooled), per channel (C = 16 or 32) --------
template <int C>
__global__ void __launch_bounds__(256)
bn_stats_kernel(const float* __restrict__ src, int rows,
                float* __restrict__ sums, float* __restrict__ sqs) {
  __shared__ float ls[C];
  __shared__ float lq[C];
  for (int i = threadIdx.x; i < C; i += blockDim.x) { ls[i] = 0.f; lq[i] = 0.f; }
  __syncthreads();
  int total = rows * C;
  for (int i = blockIdx.x * blockDim.x + threadIdx.x; i < total;
       i += gridDim.x * blockDim.x) {
    float v = fmaxf(src[i], 0.0f);          // relu; also maps -inf (empty seg) -> 0
    int c = i & (C - 1);
    atomicAdd(&ls[c], v);
    atomicAdd(&lq[c], v * v);
  }
  __syncthreads();
  for (int i = threadIdx.x; i < C; i += blockDim.x) {
    atomicAdd(&sums[i], ls[i]);
    atomicAdd(&sqs[i], lq[i]);
  }
}

__global__ void bn_finalize_kernel(const float* __restrict__ sums,
                                   const float* __restrict__ sqs,
                                   const float* __restrict__ gamma,
                                   const float* __restrict__ beta,
                                   float* __restrict__ scale,
                                   float* __restrict__ shift, int C, float invR) {
  int c = threadIdx.x;
  if (c < C) {
    float mu  = sums[c] * invR;
    float var = sqs[c] * invR - mu * mu;
    float sc  = gamma[c] * rsqrtf(var + 1e-5f);
    scale[c] = sc;
    shift[c] = beta[c] - mu * sc;
  }
}

__global__ void __launch_bounds__(256)
bn_apply1_kernel(const float* __restrict__ h1pool, const float* __restrict__ scale,
                 const float* __restrict__ shift, float* __restrict__ h1n, int total) {
  int i = blockIdx.x * blockDim.x + threadIdx.x;
  if (i >= total) return;
  int c = i & 15;
  h1n[i] = scale[c] * fmaxf(h1pool[i], 0.0f) + shift[c];
}

// ---- Layer 2: gathered GEMM [N2,432]x[432,32] on f32 WMMA, fused w/ pool ---
// Wave = 16 particles. A fragments gathered straight into registers
// (16x4 f32 A layout: lane(half,m) holds A[m][2*half], A[m][2*half+1]).
// B (= W2 reshaped [432,32]) lives in LDS interleaved by K-row pairs:
//   Bp[pair*64 + 2*n + (r&1)] = W2[r][n],  pair = r>>1
// so a B fragment {B[2p][n], B[2p+1][n]} is one contiguous 8B ds_load_b64.
__global__ void __launch_bounds__(256)
layer2_wmma_kernel(const float* __restrict__ h1n, const float* __restrict__ W2,
                   const float* __restrict__ b2, const int* __restrict__ nbr2,
                   const int* __restrict__ pool2, float* __restrict__ h2pool,
                   int n2) {
  __shared__ float W2lds[216 * 64];        // 55,296 B
  for (int i = threadIdx.x; i < 432 * 32; i += blockDim.x) {
    int r = i >> 5, c = i & 31;
    W2lds[(r >> 1) * 64 + (c << 1) + (r & 1)] = W2[i];
  }
  __syncthreads();

  const int lane = threadIdx.x & 31;
  const int wave = threadIdx.x >> 5;
  const int half = lane >> 4;              // 0: lanes 0-15, 1: lanes 16-31
  const int m    = lane & 15;
  const int tilebase = (blockIdx.x * 8 + wave) * 16;
  if (tilebase >= n2) return;              // wave-uniform exit (EXEC stays all-1)

  const int prow = min(tilebase + m, n2 - 1);
  const int* nb  = nbr2 + prow * KN;

  v8f acc0 = {};
  v8f acc1 = {};

  for (int k = 0; k < KN; ++k) {
    int g = nb[k];
    const float2* hp2 = reinterpret_cast<const float2*>(h1n + (size_t)g * 16);
    const v2f* bpair  = reinterpret_cast<const v2f*>(&W2lds[(k * 8 + half) * 64]);
    #pragma unroll
    for (int kk = 0; kk < 4; ++kk) {
      // A fragment: channels {kk*4+2h, kk*4+2h+1} of neighbor g (8B load)
      float2 aq = hp2[2 * kk + half];
      v2f a = { aq.x, aq.y };
      // B fragment: K-row pair (k*8 + kk*2 + half), columns m / m+16
      v2f b0 = bpair[kk * 64 + m];         // +kk*2 pairs = +kk*64 v2f... (v2f stride 32/pair)
      v2f b1 = bpair[kk * 64 + m + 16];
      acc0 = __builtin_amdgcn_wmma_f32_16x16x4_f32(
          false, a, false, b0, (short)0, acc0, false, false);
      acc1 = __builtin_amdgcn_wmma_f32_16x16x4_f32(
          false, a, false, b1, (short)0, acc1, false, false);
    }
  }

  // Epilogue: +bias, segment-max into h2pool. C/D layout: VGPR r ->
  // row = r + 8*half, col = m.
  float bias0 = b2[m];
  float bias1 = b2[m + 16];
  #pragma unroll
  for (int r = 0; r < 8; ++r) {
    int p = tilebase + r + half * 8;
    if (p < n2) {
      int dst = pool2[p] * 32;
      atomicMaxF(&h2pool[dst + m],      acc0[r] + bias0);
      atomicMaxF(&h2pool[dst + m + 16], acc1[r] + bias1);
    }
  }
}

// ---- Final: relu+BN2 apply, FC 32->64->10, mean-accumulate logits ----------
__global__ void __launch_bounds__(256)
final_rows_kernel(const float* __restrict__ h2pool, const float* __restrict__ scale2,
                  const float* __restrict__ shift2, const float* __restrict__ Wfc1,
                  const float* __restrict__ bfc1, const float* __restrict__ Wfc2,
                  const float* __restrict__ bfc2, float* __restrict__ logit_acc,
                  int n3) {
  __shared__ float lred[10];
  if (threadIdx.x < 10) lred[threadIdx.x] = 0.0f;
  __syncthreads();

  int nid = blockIdx.x * blockDim.x + threadIdx.x;
  if (nid < n3) {
    float h[32];
    #pragma unroll
    for (int c = 0; c < 32; ++c)
      h[c] = scale2[c] * fmaxf(h2pool[(size_t)nid * 32 + c], 0.0f) + shift2[c];

    float l[10];
    #pragma unroll
    for (int o = 0; o < 10; ++o) l[o] = bfc2[o];

    for (int j = 0; j < 64; ++j) {
      float z = bfc1[j];
      #pragma unroll
      for (int c = 0; c < 32; ++c) z = fmaf(h[c], Wfc1[c * 64 + j], z);
      #pragma unroll
      for (int o = 0; o < 10; ++o) l[o] = fmaf(z, Wfc2[j * 10 + o], l[o]);
    }
    #pragma unroll
    for (int o = 0; o < 10; ++o) atomicAdd(&lred[o], l[o]);
  }
  __syncthreads();
  if (threadIdx.x < 10) atomicAdd(&logit_acc[threadIdx.x], lred[threadIdx.x]);
}

__global__ void softmax_kernel(const float* __restrict__ logit_acc,
                               float* __restrict__ out, float invN) {
  if (threadIdx.x == 0 && blockIdx.x == 0) {
    float l[10];
    float mx = -1e30f;
    for (int o = 0; o < 10; ++o) { l[o] = logit_acc[o] * invN; mx = fmaxf(mx, l[o]); }
    float s = 0.0f;
    for (int o = 0; o < 10; ++o) { l[o] = expf(l[o] - mx); s += l[o]; }
    for (int o = 0; o < 10; ++o) out[o] = l[o] / s;
  }
}

// ---------------------------------------------------------------------------
extern "C" void kernel_launch(void* const* d_in, const int* in_sizes, int n_in,
                              void* d_out, int out_size, void* d_ws, size_t ws_size,
                              hipStream_t stream) {
  (void)in_sizes; (void)n_in; (void)out_size; (void)ws_size;

  const float* x      = (const float*)d_in[0];
  const float* W1     = (const float*)d_in[1];
  const float* b1     = (const float*)d_in[2];
  const float* gamma1 = (const float*)d_in[3];
  const float* beta1  = (const float*)d_in[4];
  const float* W2     = (const float*)d_in[5];
  const float* b2     = (const float*)d_in[6];
  const float* gamma2 = (const float*)d_in[7];
  const float* beta2  = (const float*)d_in[8];
  const float* Wfc1   = (const float*)d_in[9];
  const float* bfc1   = (const float*)d_in[10];
  const float* Wfc2   = (const float*)d_in[11];
  const float* bfc2   = (const float*)d_in[12];
  const int* nbr1     = (const int*)d_in[13];
  const int* stencil1 = (const int*)d_in[14];
  const int* pool1    = (const int*)d_in[15];
  const int* nbr2     = (const int*)d_in[16];
  const int* pool2    = (const int*)d_in[17];

  float* ws      = (float*)d_ws;
  float* h1pool  = ws;                     // N2*16
  float* h2pool  = ws + 2000000;           // N3*32
  float* stats   = ws + 2500000;           // 256
  float* h1n     = ws + 2500256;           // N2*16
  float* sums1 = stats +   0, *sqs1 = stats + 16, *scale1 = stats + 32, *shift1 = stats + 48;
  float* sums2 = stats +  64, *sqs2 = stats + 96, *scale2 = stats + 128, *shift2 = stats + 160;
  float* logit_acc = stats + 192;

  // 1. init pools to -inf, stats/logits to 0
  init_ws_kernel<<<(2500256 + 255) / 256, 256, 0, stream>>>(ws);
  // 2. layer1 conv + segment-max
  layer1_kernel<<<(N1C + 255) / 256, 256, 0, stream>>>(
      x, W1, b1, nbr1, stencil1, pool1, h1pool, N1C);
  // 3-5. BN1 stats / finalize / apply -> h1n
  bn_stats_kernel<16><<<480, 256, 0, stream>>>(h1pool, N2C, sums1, sqs1);
  bn_finalize_kernel<<<1, 32, 0, stream>>>(sums1, sqs1, gamma1, beta1,
                                           scale1, shift1, 16, 1.0f / N2C);
  bn_apply1_kernel<<<(N2C * 16 + 255) / 256, 256, 0, stream>>>(
      h1pool, scale1, shift1, h1n, N2C * 16);
  // 6. layer2 gathered GEMM on f32 WMMA + bias + segment-max
  layer2_wmma_kernel<<<(N2C + 127) / 128, 256, 0, stream>>>(
      h1n, W2, b2, nbr2, pool2, h2pool, N2C);
  // 7-8. BN2 stats / finalize
  bn_stats_kernel<32><<<480, 256, 0, stream>>>(h2pool, N3C, sums2, sqs2);
  bn_finalize_kernel<<<1, 32, 0, stream>>>(sums2, sqs2, gamma2, beta2,
                                           scale2, shift2, 32, 1.0f / N3C);
  // 9. BN2 apply + FC1 + FC2 + logit mean accumulation
  final_rows_kernel<<<(N3C + 255) / 256, 256, 0, stream>>>(
      h2pool, scale2, shift2, Wfc1, bfc1, Wfc2, bfc2, logit_acc, N3C);
  // 10. softmax -> d_out (10 floats)
  softmax_kernel<<<1, 32, 0, stream>>>(logit_acc, (float*)d_out, 1.0f / N3C);
}